// GCN_layer_18184891531604
// MI455X (gfx1250) — compile-verified
//
#include <hip/hip_runtime.h>

typedef __attribute__((ext_vector_type(2))) float v2f;
typedef __attribute__((ext_vector_type(8))) float v8f;

#define D_IN  256
#define D_OUT 128

// ---------------------------------------------------------------------------
// Phase 0: out[n][d] = b[d]   (bias broadcast; also clears poisoned d_out)
// ---------------------------------------------------------------------------
__global__ __launch_bounds__(256) void gcn_init_out(const float* __restrict__ bias,
                                                    float* __restrict__ out,
                                                    long long total4) {
    long long i = (long long)blockIdx.x * blockDim.x + threadIdx.x;
    if (i < total4) {
        int d4 = (int)(i & (D_OUT / 4 - 1));
        ((float4*)out)[i] = ((const float4*)bias)[d4];
    }
}

// ---------------------------------------------------------------------------
// Phase 1: support = X @ W  via V_WMMA_F32_16X16X4_F32 (full fp32 precision)
// One wave computes a 16(M) x 128(N) strip: 8 accumulator tiles, A read once.
// ---------------------------------------------------------------------------
__global__ __launch_bounds__(256) void gcn_gemm_xw(const float* __restrict__ X,
                                                   const float* __restrict__ W,
                                                   float* __restrict__ S,
                                                   int n_rows) {
    const int lane  = threadIdx.x & 31;
    const int wave  = threadIdx.x >> 5;
    const int mtile = blockIdx.x * 8 + wave;      // 8 waves / block
    if (mtile * 16 + 16 > n_rows) return;         // whole-wave guard (EXEC stays all-1)
    const int m0 = mtile * 16;

    // A-matrix 16x4 f32 layout: lanes 0-15 -> M=lane,   K = {0,1}
    //                           lanes16-31 -> M=lane-16,K = {2,3}
    const int am = lane & 15;
    const int ak = (lane >> 4) * 2;
    // B-matrix 4x16 f32 layout: VGPR0: lanes0-15 K=0 | lanes16-31 K=2
    //                           VGPR1: lanes0-15 K=1 | lanes16-31 K=3
    const int bn = lane & 15;
    const int bk = (lane >> 4) * 2;

    v8f c[8];
#pragma unroll
    for (int t = 0; t < 8; ++t) c[t] = (v8f){0.f, 0.f, 0.f, 0.f, 0.f, 0.f, 0.f, 0.f};

    const float* xrow = X + (long long)(m0 + am) * D_IN;

    for (int k = 0; k < D_IN; k += 4) {
        v2f a = *(const v2f*)(xrow + k + ak);     // 8-byte aligned (k even, ak even)
#pragma unroll
        for (int t = 0; t < 8; ++t) {
            v2f b;
            b.x = W[(long long)(k + bk)     * D_OUT + t * 16 + bn];
            b.y = W[(long long)(k + bk + 1) * D_OUT + t * 16 + bn];
            c[t] = __builtin_amdgcn_wmma_f32_16x16x4_f32(
                /*neg_a=*/false, a, /*neg_b=*/false, b,
                /*c_mod=*/(short)0, c[t], /*reuse_a=*/false, /*reuse_b=*/false);
        }
    }

    // C/D 16x16 f32 layout: VGPR r: lanes0-15 -> M=r, N=lane ; lanes16-31 -> M=r+8
    const int cn    = lane & 15;
    const int cmoff = (lane >> 4) * 8;
#pragma unroll
    for (int t = 0; t < 8; ++t) {
#pragma unroll
        for (int r = 0; r < 8; ++r) {
            S[(long long)(m0 + r + cmoff) * D_OUT + t * 16 + cn] = c[t][r];
        }
    }
}

// ---------------------------------------------------------------------------
// Phase 2: for each edge e: out[rows[e]] += vals[e] * support[cols[e]]
// One wave per edge: 32 lanes x float4 = 128 floats.
//  - edge index forced wave-uniform (readfirstlane) -> rows/cols/vals become
//    scalar s_load_b32 (KMcnt), gather becomes SGPR-base global_load_b128
//  - scatter uses guaranteed hardware fp32 atomics via inline asm
//    (global_atomic_add_f32, no return -> STOREcnt; resolves in L2 since
//    out (51.2 MB) fits the 192 MB L2)
// ---------------------------------------------------------------------------
__global__ __launch_bounds__(256) void gcn_edge_scatter(const int* __restrict__ rows,
                                                        const int* __restrict__ cols,
                                                        const float* __restrict__ vals,
                                                        const float* __restrict__ S,
                                                        float* __restrict__ out,
                                                        int n_edges) {
    const int lane = threadIdx.x & 31;
    int e = (int)(blockIdx.x * 8u) + (threadIdx.x >> 5);
    e = __builtin_amdgcn_readfirstlane(e);        // provably wave-uniform -> SGPR
    if (e >= n_edges) return;                     // scalar branch, EXEC untouched

    const int   r = rows[e];                      // s_load_b32
    const int   c = cols[e];                      // s_load_b32
    const float v = vals[e];                      // s_load_b32

    const float4 s = *(const float4*)(S + (long long)c * D_OUT + lane * 4);

    float4 g;
    g.x = s.x * v;
    g.y = s.y * v;
    g.z = s.z * v;
    g.w = s.w * v;

    float* dst = out + (long long)r * D_OUT + lane * 4;
    asm volatile("global_atomic_add_f32 %0, %1, off"           :: "v"(dst), "v"(g.x) : "memory");
    asm volatile("global_atomic_add_f32 %0, %1, off offset:4"  :: "v"(dst), "v"(g.y) : "memory");
    asm volatile("global_atomic_add_f32 %0, %1, off offset:8"  :: "v"(dst), "v"(g.z) : "memory");
    asm volatile("global_atomic_add_f32 %0, %1, off offset:12" :: "v"(dst), "v"(g.w) : "memory");
}

// ---------------------------------------------------------------------------
// Launch. Inputs: rows[E], cols[E], vals[E], x[N,256], W[256,128], b[128], n_nodes
// ---------------------------------------------------------------------------
extern "C" void kernel_launch(void* const* d_in, const int* in_sizes, int n_in,
                              void* d_out, int out_size, void* d_ws, size_t ws_size,
                              hipStream_t stream) {
    const int*   rows = (const int*)d_in[0];
    const int*   cols = (const int*)d_in[1];
    const float* vals = (const float*)d_in[2];
    const float* x    = (const float*)d_in[3];
    const float* W    = (const float*)d_in[4];
    const float* bias = (const float*)d_in[5];

    const int E = in_sizes[0];
    const int N = in_sizes[3] / D_IN;      // 100000 (multiple of 16)

    float* support = (float*)d_ws;         // N * 128 floats = 51.2 MB scratch

    // Phase 0: bias broadcast / zero-init of out
    {
        long long total4 = (long long)N * (D_OUT / 4);
        int blocks = (int)((total4 + 255) / 256);
        gcn_init_out<<<blocks, 256, 0, stream>>>(bias, (float*)d_out, total4);
    }

    // Phase 1: WMMA GEMM -> support
    {
        int mtiles = N / 16;               // exact for N=100000
        int blocks = (mtiles + 7) / 8;     // 8 waves (strips) per 256-thread block
        gcn_gemm_xw<<<blocks, 256, 0, stream>>>(x, W, support, N);
    }

    // Phase 2: edge gather/scale/scatter-add
    {
        int blocks = (E + 7) / 8;          // 8 edges (waves) per block
        gcn_edge_scatter<<<blocks, 256, 0, stream>>>(rows, cols, vals, support,
                                                     (float*)d_out, E);
    }
}